// ConvDemodulated_77721728188932
// MI455X (gfx1250) — compile-verified
//
#include <hip/hip_runtime.h>

typedef __attribute__((ext_vector_type(2))) float v2f;
typedef __attribute__((ext_vector_type(8))) float v8f;

#define HW_PIX 65536   // 256*256
#define C_IN   64
#define C_OUT  64
#define NBATCH 16

// Step 1: demodulate weights: wn[o][i] = w[o][i] * rsqrt(sum_i w[o][i]^2 + 1e-8)
__global__ void wnorm_kernel(const float* __restrict__ w, float* __restrict__ wn) {
    int o = threadIdx.x;            // 64 threads, one output channel each
    float s = 1e-8f;
#pragma unroll
    for (int i = 0; i < C_IN; ++i) {
        float v = w[o * C_IN + i];
        s += v * v;
    }
    float r = rsqrtf(s);
#pragma unroll
    for (int i = 0; i < C_IN; ++i) {
        wn[o * C_IN + i] = w[o * C_IN + i] * r;
    }
}

// Step 2: per-batch GEMM Out = Wn(64x64) @ X(64x65536), bias add, clamp.
// Block = 256 threads = 8 waves. Each wave: 64(M) x 16(N) tile via
// V_WMMA_F32_16X16X4_F32, K=64 in steps of 4.
__global__ __launch_bounds__(256, 2) void conv1x1_wmma_kernel(
    const float* __restrict__ x,     // (B, C_IN, H, W)
    const float* __restrict__ wn,    // (C_OUT, C_IN) demodulated
    const float* __restrict__ bias,  // (C_OUT)
    float* __restrict__ out)         // (B, C_OUT, H, W)
{
    __shared__ float lds_w[C_OUT * C_IN];   // 16 KB, reused by all waves/tiles

    // cooperative weight stage: 4096 floats / 256 threads
    for (int i = threadIdx.x; i < C_OUT * C_IN; i += 256)
        lds_w[i] = wn[i];
    __syncthreads();

    const int lane = threadIdx.x & 31;
    const int wave = threadIdx.x >> 5;
    const int b    = blockIdx.y;
    const int col  = blockIdx.x * 128 + wave * 16 + (lane & 15); // pixel column
    const int half = lane >> 4;          // 0: lanes 0-15, 1: lanes 16-31
    const int koff = half * 2;           // A/B VGPR K sub-row per ISA layout
    const int moff = half * 8;           // C/D row offset per ISA layout
    const int m0   = lane & 15;          // A-matrix row index

    const float* xb = x + ((size_t)b * C_IN) * HW_PIX + col;

    v8f acc0 = {}, acc1 = {}, acc2 = {}, acc3 = {};

#pragma unroll
    for (int k = 0; k < C_IN; k += 4) {
        // B tile (4x16 f32): lane holds column (lane&15); VGPR0 = row k+koff,
        // VGPR1 = row k+koff+1 (rows k+2,k+3 live in lanes 16-31 via koff).
        v2f bm;
        bm.x = xb[(size_t)(k + koff) * HW_PIX];
        bm.y = xb[(size_t)(k + koff + 1) * HW_PIX];

        // A tiles (16x4 f32) from LDS: row m in lane m, K-pair selected by koff.
        // Consecutive K values -> single 8-byte ds_load.
        v2f a0 = *(const v2f*)&lds_w[(m0 +  0) * C_IN + k + koff];
        v2f a1 = *(const v2f*)&lds_w[(m0 + 16) * C_IN + k + koff];
        v2f a2 = *(const v2f*)&lds_w[(m0 + 32) * C_IN + k + koff];
        v2f a3 = *(const v2f*)&lds_w[(m0 + 48) * C_IN + k + koff];

        acc0 = __builtin_amdgcn_wmma_f32_16x16x4_f32(false, a0, false, bm, (short)0, acc0, false, false);
        acc1 = __builtin_amdgcn_wmma_f32_16x16x4_f32(false, a1, false, bm, (short)0, acc1, false, false);
        acc2 = __builtin_amdgcn_wmma_f32_16x16x4_f32(false, a2, false, bm, (short)0, acc2, false, false);
        acc3 = __builtin_amdgcn_wmma_f32_16x16x4_f32(false, a3, false, bm, (short)0, acc3, false, false);
    }

    // Epilogue: C/D layout = VGPR r holds rows (moff+r) for lanes' half,
    // N = lane&15. Add bias, clamp to +-256, store coalesced per row.
    float* ob = out + ((size_t)b * C_OUT) * HW_PIX + col;

#pragma unroll
    for (int r = 0; r < 8; ++r) {
        {
            int m = 0 + moff + r;
            float v = acc0[r] + bias[m];
            v = fminf(fmaxf(v, -256.0f), 256.0f);
            ob[(size_t)m * HW_PIX] = v;
        }
        {
            int m = 16 + moff + r;
            float v = acc1[r] + bias[m];
            v = fminf(fmaxf(v, -256.0f), 256.0f);
            ob[(size_t)m * HW_PIX] = v;
        }
        {
            int m = 32 + moff + r;
            float v = acc2[r] + bias[m];
            v = fminf(fmaxf(v, -256.0f), 256.0f);
            ob[(size_t)m * HW_PIX] = v;
        }
        {
            int m = 48 + moff + r;
            float v = acc3[r] + bias[m];
            v = fminf(fmaxf(v, -256.0f), 256.0f);
            ob[(size_t)m * HW_PIX] = v;
        }
    }
}

extern "C" void kernel_launch(void* const* d_in, const int* in_sizes, int n_in,
                              void* d_out, int out_size, void* d_ws, size_t ws_size,
                              hipStream_t stream) {
    const float* x    = (const float*)d_in[0];   // (16,64,256,256) f32
    const float* w    = (const float*)d_in[1];   // (64,64,1,1) f32
    const float* bias = (const float*)d_in[2];   // (64,) f32
    float*       out  = (float*)d_out;           // (16,64,256,256) f32
    float*       wn   = (float*)d_ws;            // 64*64 floats scratch

    wnorm_kernel<<<dim3(1), dim3(64), 0, stream>>>(w, wn);

    // 65536 cols / (8 waves * 16 cols) = 512 column tiles; y = batch
    dim3 grid(HW_PIX / 128, NBATCH);
    conv1x1_wmma_kernel<<<grid, dim3(256), 0, stream>>>(x, wn, bias, out);
}